// EWMedian_14396730377050
// MI455X (gfx1250) — compile-verified
//
#include <hip/hip_runtime.h>
#include <cstdint>

#ifndef __has_builtin
#define __has_builtin(x) 0
#endif

// ---- problem constants (from reference) ------------------------------------
namespace {
constexpr int BB    = 128;
constexpr int TT    = 2048;
constexpr int CC    = 96;
constexpr int LANES = 32;                       // wave32: one 32-col slice per wave
constexpr int CH    = 32;                       // rows staged per LDS chunk
constexpr int NCH   = (TT - 1 + CH - 1) / CH;   // 64 chunks cover rows 1..2047
constexpr int TAILR = (TT - 1) - (NCH - 1) * CH; // 31 rows in the last chunk

constexpr double TEMP_D = 0.01 + 1e-8;
constexpr float  KHALF  = (float)(1.0 / (2.0 * TEMP_D));   // 1/(2*temp)
constexpr float  CHALF  = 0.025f;                          // eta*0.5  (tau == 0.5)
constexpr float  KC     = (float)((1.0 / (2.0 * TEMP_D)) * 0.025);  // KHALF*CHALF
}

// v_tanh_f32 is a gfx1250 TRANS op. Prefer the builtin (compiler manages the
// TRANS data hazard); fall back to inline asm with the required trailing
// independent op (V_NOP, ISA §7.4 item 3).
__device__ __forceinline__ float fast_tanh(float z) {
#if __has_builtin(__builtin_amdgcn_tanhf)
  return __builtin_amdgcn_tanhf(z);
#elif defined(__AMDGCN__)
  float r;
  asm volatile("v_tanh_f32 %0, %1\n\tv_nop" : "=v"(r) : "v"(z));
  return r;
#else
  return tanhf(z);
#endif
}

// Async DMA of 16B/lane from global into LDS (ASYNCcnt-tracked, CDNA5).
__device__ __forceinline__ void async_b128(unsigned lds_off, const float* g) {
#if defined(__AMDGCN__)
  asm volatile("global_load_async_to_lds_b128 %0, %1, off"
               :: "v"(lds_off), "v"((unsigned long long)(uintptr_t)g)
               : "memory");
#endif
}

__device__ __forceinline__ void wait_async_le8() {
#if defined(__AMDGCN__)
  asm volatile("s_wait_asynccnt 0x8" ::: "memory");
#endif
}
__device__ __forceinline__ void wait_async_0() {
#if defined(__AMDGCN__)
  asm volatile("s_wait_asynccnt 0x0" ::: "memory");
#endif
}

// z-space recurrence over one staged chunk; ROWS is compile-time so the body
// is fully unrolled and branch-free. Critical path per step: v_tanh -> v_fma;
// the (z + d) add, the m-update fma, the ds_load and the store all sit in the
// tanh latency shadow.
template <int ROWS>
__device__ __forceinline__ void compute_chunk(const float* __restrict__ cur,
                                              float* __restrict__ o,
                                              int lane, float& m) {
  float xkr = cur[lane] * KHALF;
  float z   = fmaf(KHALF, m, -xkr);
#pragma unroll
  for (int r = 0; r < ROWS; ++r) {
    const float t = fast_tanh(z);
    m = fmaf(-CHALF, t, m);
    o[(size_t)r * CC] = m;                      // 128B coalesced per wave
    if (r + 1 < ROWS) {                         // compile-time folded
      const float xkn = cur[(r + 1) * LANES + lane] * KHALF;
      z   = fmaf(-KC, t, z + (xkr - xkn));
      xkr = xkn;
    }
  }
}

__global__ __launch_bounds__(LANES)
void ewmedian_scan_kernel(const float* __restrict__ x, float* __restrict__ out) {
  // double-buffered LDS stage: 2 x (32 rows x 32 cols) floats = 8 KB
  __shared__ float sbuf[2][CH * LANES];

  const int lane = threadIdx.x;                 // 0..31
  const int b    = (int)blockIdx.x / (CC / LANES);
  const int cgrp = (int)blockIdx.x % (CC / LANES);

  const size_t base = (size_t)b * TT * CC;
  const float* xb   = x   + base + cgrp * LANES;          // 32-col slice, row 0
  float*       ob   = out + base + cgrp * LANES + lane;

  // LDS byte offsets (flat LDS address low 32 bits == wave-relative LDS addr)
  const unsigned lds0 = (unsigned)(uintptr_t)(&sbuf[0][0]);
  const unsigned lds1 = (unsigned)(uintptr_t)(&sbuf[1][0]);

  // m0 = x[b,0,c]; first output row.
  float m = xb[lane];
  ob[0] = m;

  const int sub   = lane & 7;    // which 16B unit in the 128B LDS row
  const int rbase = lane >> 3;   // 0..3

  // prologue: stage chunk 0 (rows 1..32)
  {
    const float* g0 = xb + (size_t)CC;          // row 1
#pragma unroll
    for (int i = 0; i < 8; ++i) {
      const int rloc = i * 4 + rbase;
      async_b128(lds0 + (unsigned)(rloc * 128 + sub * 16),
                 g0 + (size_t)rloc * CC + sub * 4);
    }
  }

  for (int k = 0; k < NCH; ++k) {
    const int r0 = 1 + k * CH;

    // stage chunk k+1 into the other buffer, then wait for chunk k (in-order)
    if (k + 1 < NCH) {
      const int      nr0  = r0 + CH;
      const int      maxr = (TT - 1) - nr0;     // clamp tail rows (stay in-bounds)
      const float*   gn   = xb + (size_t)nr0 * CC;
      const unsigned nl   = (k & 1) ? lds0 : lds1;
#pragma unroll
      for (int i = 0; i < 8; ++i) {
        const int rloc = i * 4 + rbase;
        const int rg   = rloc > maxr ? maxr : rloc;
        async_b128(nl + (unsigned)(rloc * 128 + sub * 16),
                   gn + (size_t)rg * CC + sub * 4);
      }
      wait_async_le8();    // <=8 outstanding => chunk k's 8 loads complete
    } else {
      wait_async_0();      // final chunk: drain
    }

    const float* cur = sbuf[k & 1];             // ds_load, bank-conflict-free
    float* o = ob + (size_t)r0 * CC;

    if (k + 1 < NCH) compute_chunk<CH>(cur, o, lane, m);     // 63 full chunks
    else             compute_chunk<TAILR>(cur, o, lane, m);  // 31-row tail
  }
}

extern "C" void kernel_launch(void* const* d_in, const int* /*in_sizes*/, int /*n_in*/,
                              void* d_out, int /*out_size*/, void* /*d_ws*/, size_t /*ws_size*/,
                              hipStream_t stream) {
  const float* x  = (const float*)d_in[0];
  float* out      = (float*)d_out;
  const int grid  = BB * (CC / LANES);          // 384 single-wave workgroups
  ewmedian_scan_kernel<<<grid, LANES, 0, stream>>>(x, out);
}